// GATLayer_4432406249498
// MI455X (gfx1250) — compile-verified
//
#include <hip/hip_runtime.h>

#define N_ROWS 8192
#define IN_FEAT 256
#define OUT_FEAT 128
#define SLOPE 0.1f
#define LOG2E 1.4426950408889634f

typedef __attribute__((ext_vector_type(2))) float v2f;
typedef __attribute__((ext_vector_type(8))) float v8f;

__device__ __forceinline__ float lrelu(float x) { return fmaxf(x, SLOPE * x); }
__device__ __forceinline__ float fast_exp2(float x) {
  return __builtin_amdgcn_exp2f(x);  // v_exp_f32
}

// ---------------------------------------------------------------------------
// Kernel 1: h = doc @ W_w^T + W_b   (8192x256 @ 256x128 -> 8192x128, fp32 WMMA)
// One wave per 16x16 output tile, K=256 in steps of 4.
// Output layout is "paired": hp[(j>>1)*128 + n] = float2{ h[j][n], h[j+1][n] }
// which is exactly the per-lane B-operand of V_WMMA_F32_16X16X4_F32.
// ---------------------------------------------------------------------------
__global__ __launch_bounds__(32) void k_gemm_h(const float* __restrict__ doc,
                                               const float* __restrict__ Ww,
                                               const float* __restrict__ Wb,
                                               float* __restrict__ hp) {
  const int m0 = blockIdx.x * 16;
  const int n0 = blockIdx.y * 16;
  const int lane = threadIdx.x;
  const int r = lane & 15;
  const int half = lane >> 4;

  v8f c = {0.f, 0.f, 0.f, 0.f, 0.f, 0.f, 0.f, 0.f};
  // A: doc rows (M = m0+r), K pair at k + 2*half
  const float* arow = doc + (size_t)(m0 + r) * IN_FEAT + half * 2;
  // B: (W_w^T)[k][n] = Ww[n][k]; lane holds column n = n0+r, K pair at k+2*half
  const float* brow = Ww + (size_t)(n0 + r) * IN_FEAT + half * 2;

  for (int k = 0; k < IN_FEAT; k += 4) {
    v2f a = *(const v2f*)(arow + k);
    v2f b = *(const v2f*)(brow + k);
    c = __builtin_amdgcn_wmma_f32_16x16x4_f32(false, a, false, b, (short)0, c,
                                              false, false);
  }
  const float bias = Wb[n0 + r];
#pragma unroll
  for (int i = 0; i < 8; ++i) {
    const int row = m0 + i + half * 8;  // (row & 1) == (i & 1)
    hp[((size_t)(row >> 1) * OUT_FEAT + n0 + r) * 2 + (i & 1)] = c[i] + bias;
  }
}

// ---------------------------------------------------------------------------
// Kernel 2: s_src = h @ a_src, s_dst = h @ a_dst   (one thread per row)
// Reads the paired layout: h[row][f] = hp[((row>>1)*128 + f)*2 + (row&1)]
// ---------------------------------------------------------------------------
__global__ __launch_bounds__(256) void k_sdots(const float* __restrict__ hp,
                                               const float* __restrict__ aw,
                                               float* __restrict__ ssrc,
                                               float* __restrict__ sdst) {
  const int row = blockIdx.x * 256 + threadIdx.x;
  if (row >= N_ROWS) return;
  const float* hr = hp + (size_t)(row >> 1) * OUT_FEAT * 2 + (row & 1);
  float as = 0.f, ad = 0.f;
  for (int f = 0; f < OUT_FEAT; ++f) {
    const float v = hr[2 * f];
    as = fmaf(v, aw[f], as);
    ad = fmaf(v, aw[OUT_FEAT + f], ad);
  }
  ssrc[row] = as;
  sdst[row] = ad;
}

// ---------------------------------------------------------------------------
// Kernel 3: top-2 max (+argmax) of s_dst  (single block)
// ---------------------------------------------------------------------------
__global__ __launch_bounds__(256) void k_dmax(const float* __restrict__ sdst,
                                              float* __restrict__ stats) {
  __shared__ float m1s[256], m2s[256];
  __shared__ int i1s[256];
  const int tid = threadIdx.x;
  float m1 = -3.4e38f, m2 = -3.4e38f;
  int i1 = -1;
  for (int j = tid; j < N_ROWS; j += 256) {
    const float v = sdst[j];
    if (v > m1) { m2 = m1; m1 = v; i1 = j; }
    else if (v > m2) { m2 = v; }
  }
  m1s[tid] = m1; m2s[tid] = m2; i1s[tid] = i1;
  __syncthreads();
  for (int s = 128; s > 0; s >>= 1) {
    if (tid < s) {
      const float am1 = m1s[tid], am2 = m2s[tid];
      const float bm1 = m1s[tid + s], bm2 = m2s[tid + s];
      const int bi1 = i1s[tid + s];
      if (bm1 > am1) {
        m1s[tid] = bm1; i1s[tid] = bi1; m2s[tid] = fmaxf(am1, bm2);
      } else {
        m2s[tid] = fmaxf(am2, bm1);
      }
    }
    __syncthreads();
  }
  if (tid == 0) {
    stats[0] = m1s[0];
    stats[1] = m2s[0];
    ((int*)stats)[2] = i1s[0];
  }
}

// ---------------------------------------------------------------------------
// Kernel 4: fused softmax row-block + att write + att@h via fp32 WMMA.
// One block (8 wave32) per 16 attention rows. s_dst cached (log2e-scaled) in
// LDS; exponent evaluated with v_exp_f32 using lrelu's positive homogeneity.
// ---------------------------------------------------------------------------
__global__ __launch_bounds__(256) void k_gat(const float* __restrict__ hp,
                                             const float* __restrict__ ssrc,
                                             const float* __restrict__ sdst_g,
                                             const float* __restrict__ stats,
                                             const float* __restrict__ ab_p,
                                             const int* __restrict__ selfLink_p,
                                             float* __restrict__ att,
                                             float* __restrict__ hout) {
  __shared__ __align__(16) float sdst[N_ROWS];  // pre-scaled by log2(e)
  __shared__ float red[256];
  __shared__ float creduce[16 * OUT_FEAT];
  __shared__ float srcAB[16], rowmax2[16], invden[16];

  const int tid = threadIdx.x;
  const int row0 = blockIdx.x * 16;
  const float ab = ab_p[0];
  const int self0 = (selfLink_p[0] == 0);

  for (int i = tid; i < N_ROWS / 4; i += 256) {
    float4 v = ((const float4*)sdst_g)[i];
    v.x *= LOG2E; v.y *= LOG2E; v.z *= LOG2E; v.w *= LOG2E;
    ((float4*)sdst)[i] = v;
  }
  for (int i = tid; i < 16 * OUT_FEAT; i += 256) creduce[i] = 0.f;
  if (tid < 16) {
    const float ss = ssrc[row0 + tid];
    const float dmax = stats[0], dmax2 = stats[1];
    const int amax = ((const int*)stats)[2];
    const float dm = (self0 && amax == row0 + tid) ? dmax2 : dmax;
    srcAB[tid] = (ss + ab) * LOG2E;
    // exact row max (leaky_relu is monotone); scaled into log2 domain
    rowmax2[tid] = lrelu(ss + dm + ab) * LOG2E;
  }
  __syncthreads();

  // ---- Pass A: softmax denominators (each (row, j%16) pair handled once) ---
  {
    const int r = tid & 15;
    const int p = tid >> 4;
    const float sa = srcAB[r];
    const float rm = rowmax2[r];
    float acc = 0.f;
    for (int j = p; j < N_ROWS; j += 16) {
      const float t = sa + sdst[j];
      float e = fast_exp2(fmaxf(t, SLOPE * t) - rm);
      if (self0 && j == row0 + r) e = 0.f;
      acc += e;
    }
    red[r * 16 + p] = acc;
  }
  __syncthreads();
  if (tid < 16) {
    float s = 0.f;
#pragma unroll
    for (int p = 0; p < 16; ++p) s += red[tid * 16 + p];
    invden[tid] = 1.f / s;
  }
  __syncthreads();

  // ---- Pass B: write normalized att + accumulate att@h with WMMA ----------
  {
    const int wave = tid >> 5;
    const int lane = tid & 31;
    const int r = lane & 15;
    const int half = lane >> 4;
    const float sa = srcAB[r];
    const float rm = rowmax2[r];
    const float inv = invden[r];
    const int iglob = row0 + r;
    float* attrow = att + (size_t)iglob * N_ROWS;
    const float2* hpv = (const float2*)hp;

    const v8f zero = {0.f, 0.f, 0.f, 0.f, 0.f, 0.f, 0.f, 0.f};
    v8f c[8];
#pragma unroll
    for (int f = 0; f < 8; ++f) c[f] = zero;

    const int jbeg = wave * (N_ROWS / 8);
    const int jend = jbeg + (N_ROWS / 8);
    for (int j0 = jbeg; j0 < jend; j0 += 4) {
      const int jA = j0 + half * 2;  // this lane's K-pair columns
      const float t0 = sa + sdst[jA];
      const float t1 = sa + sdst[jA + 1];
      float e0 = fast_exp2(fmaxf(t0, SLOPE * t0) - rm) * inv;
      float e1 = fast_exp2(fmaxf(t1, SLOPE * t1) - rm) * inv;
      if (self0) {
        if (jA == iglob) e0 = 0.f;
        if (jA + 1 == iglob) e1 = 0.f;
      }
      *(float2*)(attrow + jA) = make_float2(e0, e1);  // normalized att out

      v2f a = {e0, e1};
      // B operand pair {h[jA][n], h[jA+1][n]} = one b64 load from paired hp
      const float2* hb = hpv + (size_t)(jA >> 1) * OUT_FEAT + r;
#pragma unroll
      for (int f = 0; f < 8; ++f) {
        const float2 p = hb[f * 16];
        v2f b = {p.x, p.y};
        c[f] = __builtin_amdgcn_wmma_f32_16x16x4_f32(false, a, false, b,
                                                     (short)0, c[f], false,
                                                     false);
      }
    }
    // split-K reduction across the 8 waves via LDS float atomics
#pragma unroll
    for (int f = 0; f < 8; ++f) {
#pragma unroll
      for (int i = 0; i < 8; ++i) {
        const int m = i + half * 8;
        atomicAdd(&creduce[m * OUT_FEAT + f * 16 + r], c[f][i]);
      }
    }
  }
  __syncthreads();

  for (int i = tid; i < 16 * OUT_FEAT; i += 256) {
    const int m = i / OUT_FEAT;
    hout[(size_t)(row0 + m) * OUT_FEAT + (i % OUT_FEAT)] =
        lrelu(creduce[i] * invden[m]);
  }
}

// ---------------------------------------------------------------------------
extern "C" void kernel_launch(void* const* d_in, const int* in_sizes, int n_in,
                              void* d_out, int out_size, void* d_ws,
                              size_t ws_size, hipStream_t stream) {
  (void)in_sizes; (void)n_in; (void)out_size; (void)ws_size;
  const float* doc = (const float*)d_in[0];
  const float* Ww = (const float*)d_in[1];
  const float* Wb = (const float*)d_in[2];
  const float* aw = (const float*)d_in[3];
  const float* ab = (const float*)d_in[4];
  const int* selfLink = (const int*)d_in[5];

  float* ws = (float*)d_ws;
  float* hp = ws;                                   // 8192*128 f32 (paired)
  float* ssrc = ws + (size_t)N_ROWS * OUT_FEAT;     // 8192 f32
  float* sdst = ssrc + N_ROWS;                      // 8192 f32
  float* stats = sdst + N_ROWS;                     // max, max2, argmax(int)

  float* out = (float*)d_out;
  float* hout = out;                                // 8192*128
  float* att = out + (size_t)N_ROWS * OUT_FEAT;     // 8192*8192

  k_gemm_h<<<dim3(N_ROWS / 16, OUT_FEAT / 16), 32, 0, stream>>>(doc, Ww, Wb,
                                                                hp);
  k_sdots<<<N_ROWS / 256, 256, 0, stream>>>(hp, aw, ssrc, sdst);
  k_dmax<<<1, 256, 0, stream>>>(sdst, stats);
  k_gat<<<N_ROWS / 16, 256, 0, stream>>>(hp, ssrc, sdst, stats, ab, selfLink,
                                         att, hout);
}